// MCPBRNN_Generic_PETconstraint_Scaling_BYPASSM1_10196252360840
// MI455X (gfx1250) — compile-verified
//
#include <hip/hip_runtime.h>
#include <stdint.h>
#include <math.h>

// Problem constants from the reference
#define NB      1000000
#define SPIN_N  1000
#define TRAIN_N 800000
#define ML_C    2.9086f
#define SL_C    1.898f
#define U1_MAXC 221.519f
#define LOG2E_C 1.4426950408889634f

// -------------------------------------------------------------------------
// Folded scalar constants (H==1 -> every weight matrix is a scalar).
// sigmoid(a) = 1 / (1 + exp2(-a*log2e)); we fold -log2e into the k* terms.
// -------------------------------------------------------------------------
struct Consts {
  float kib_c, kib_u, kib_0;   // ib  = 1/(1+exp2(kib_c*c + kib_u*u1 + kib_0))
  float koo_c, koo_0;          // oo  = oo1/(1+exp2(koo_c*c + koo_0))
  float kol_u, kol_0;          // ol  = ol1/(1+exp2(kol_u*u2 + kol_0))
  float oo1, ol1;
  float mo, so;
};

__device__ __forceinline__ Consts make_consts(
    const float* cmean, const float* cstd,
    const float* wr_yom, const float* wr_ylm, const float* wr_yfm,
    const float* b0_yom, const float* w1_yom,
    const float* b0_ylm, const float* w2_ylm,
    const float* w1_yum, const float* b0_yum)
{
  Consts k;
  k.mo = cmean[0];
  k.so = cstd[0];
  float eo = expf(wr_yom[0]);
  float el = expf(wr_ylm[0]);
  float ef = expf(wr_yfm[0]);
  float inv = 1.0f / (eo + el + ef);
  k.oo1 = eo * inv;
  k.ol1 = el * inv;

  float wib = w1_yum[0];
  k.kib_c = -LOG2E_C * wib / k.so;
  k.kib_u = -LOG2E_C * wib / U1_MAXC;
  k.kib_0 = -LOG2E_C * (b0_yum[0] - k.mo * wib / k.so);

  float woo = w1_yom[0];
  k.koo_c = -LOG2E_C * woo / k.so;
  k.koo_0 = -LOG2E_C * (b0_yom[0] - k.mo * woo / k.so);

  float wol = w2_ylm[0];
  k.kol_u = -LOG2E_C * wol / SL_C;
  k.kol_0 = -LOG2E_C * (b0_ylm[0] - ML_C * wol / SL_C);
  return k;
}

// -------------------------------------------------------------------------
// Kernel A: per-block partial sum / sum-of-squares of y_obs[SPIN:TRAIN]
// -------------------------------------------------------------------------
__global__ __launch_bounds__(256) void reduce_partial(
    const float* __restrict__ y, float* __restrict__ psum, float* __restrict__ psq)
{
  __shared__ float ss[256], sq[256];
  const int tid = threadIdx.x;
  float s = 0.0f, q = 0.0f;
  for (int i = SPIN_N + blockIdx.x * 256 + tid; i < TRAIN_N; i += 1024 * 256) {
    float v = y[i];
    s += v;
    q = fmaf(v, v, q);
  }
  ss[tid] = s; sq[tid] = q;
  __syncthreads();
  for (int o = 128; o > 0; o >>= 1) {
    if (tid < o) { ss[tid] += ss[tid + o]; sq[tid] += sq[tid + o]; }
    __syncthreads();
  }
  if (tid == 0) { psum[blockIdx.x] = ss[0]; psq[blockIdx.x] = sq[0]; }
}

// -------------------------------------------------------------------------
// Kernel B: final reduction -> obsstd = std(y_obs[SPIN:TRAIN], ddof=1)
// -------------------------------------------------------------------------
__global__ __launch_bounds__(1024) void reduce_final(
    const float* __restrict__ psum, const float* __restrict__ psq, float* __restrict__ stdp)
{
  __shared__ float ss[1024], sq[1024];
  const int t = threadIdx.x;
  ss[t] = psum[t]; sq[t] = psq[t];
  __syncthreads();
  for (int o = 512; o > 0; o >>= 1) {
    if (t < o) { ss[t] += ss[t + o]; sq[t] += sq[t + o]; }
    __syncthreads();
  }
  if (t == 0) {
    const float n = (float)(TRAIN_N - SPIN_N);
    float mean = ss[0] / n;
    float var  = fmaxf(0.0f, (sq[0] - n * mean * mean) / (n - 1.0f));
    stdp[0] = sqrtf(var);
  }
}

// -------------------------------------------------------------------------
// Kernel S: single-wave sequential scan of the carry c_t.
// All 32 lanes cooperatively stage the next 32 timesteps (u1,u2 pairs, 8B
// per lane) into an LDS double-buffer via global_load_async_to_lds_b64
// (ASYNCcnt-tracked, CDNA5 async path), synchronized with s_wait_asynccnt.
// Lane 0 then runs 32 serial steps reading inputs from LDS (low latency).
// NB % 32 == 0, so no tail guards are required.
// -------------------------------------------------------------------------
__global__ __launch_bounds__(32) void scan_kernel(
    const float* __restrict__ x, float* __restrict__ cbuf,
    const float* cmean, const float* cstd,
    const float* wr_yom, const float* wr_ylm, const float* wr_yfm,
    const float* b0_yom, const float* w1_yom,
    const float* b0_ylm, const float* w2_ylm,
    const float* w1_yum, const float* b0_yum)
{
  const Consts K = make_consts(cmean, cstd, wr_yom, wr_ylm, wr_yfm,
                               b0_yom, w1_yom, b0_ylm, w2_ylm, w1_yum, b0_yum);
  const int lane = threadIdx.x;
  __shared__ float xs[2][64];   // [buffer][32 steps * 2 floats]

  // Formally initialize LDS (avoids any uninitialized-read reasoning).
  xs[0][2 * lane] = 0.0f; xs[0][2 * lane + 1] = 0.0f;
  xs[1][2 * lane] = 0.0f; xs[1][2 * lane + 1] = 0.0f;

  const int NCH = NB / 32;

  // Prologue: async-stage chunk 0 into buffer 0.
  {
    unsigned lds = (unsigned)(uintptr_t)(const void*)&xs[0][2 * lane];
    unsigned long long ga = (unsigned long long)(uintptr_t)(x + 2 * lane);
    asm volatile("global_load_async_to_lds_b64 %0, %1, off"
                 :: "v"(lds), "v"(ga) : "memory");
  }

  float c = 0.0f;
  for (int k = 0; k < NCH; ++k) {
    if (k + 1 < NCH) {
      // Stage chunk k+1 into the other buffer while chunk k is consumed.
      unsigned lds = (unsigned)(uintptr_t)(const void*)&xs[(k + 1) & 1][2 * lane];
      unsigned long long ga =
          (unsigned long long)(uintptr_t)(x + 2 * ((k + 1) * 32 + lane));
      asm volatile("global_load_async_to_lds_b64 %0, %1, off"
                   :: "v"(lds), "v"(ga) : "memory");
      asm volatile("s_wait_asynccnt 1" ::: "memory");  // chunk k landed in LDS
    } else {
      asm volatile("s_wait_asynccnt 0" ::: "memory");  // last chunk landed
    }

    if (lane == 0) {
      const float* bb = &xs[k & 1][0];
      const int base = k * 32;
      for (int j = 0; j < 32; ++j) {
        const float u1 = bb[2 * j];
        const float u2 = bb[2 * j + 1];
        cbuf[base + j] = c;  // carry ENTERING step t (the reference's c0 output)

        // Critical path: fma -> v_exp_f32 -> add -> v_rcp_f32 -> fma tree.
        float ib = __builtin_amdgcn_rcpf(
            1.0f + exp2f(fmaf(c, K.kib_c, fmaf(u1, K.kib_u, K.kib_0))));
        float oo = K.oo1 * __builtin_amdgcn_rcpf(
            1.0f + exp2f(fmaf(c, K.koo_c, K.koo_0)));
        float ol = K.ol1 * __builtin_amdgcn_rcpf(
            1.0f + exp2f(fmaf(u2, K.kol_u, K.kol_0)));
        // ol - relu(ol - u2/c) == min(ol, u2/c) for c > 0
        float olc = (c > 0.0f) ? fminf(ol, u2 * __builtin_amdgcn_rcpf(c)) : ol;
        float f = 1.0f - oo - olc;
        c = fmaf(f, c, (1.0f - ib) * u1);
      }
    }
  }
}

// -------------------------------------------------------------------------
// Kernel E: embarrassingly parallel epilogue. Recompute gates from (c_t,u_t),
// apply time_lag mask, write all 13*NB output floats.
// -------------------------------------------------------------------------
__global__ __launch_bounds__(256) void out_kernel(
    const float* __restrict__ x, const float* __restrict__ cbuf,
    const float* __restrict__ stdp, const int* __restrict__ time_lag,
    float* __restrict__ out,
    const float* cmean, const float* cstd,
    const float* wr_yom, const float* wr_ylm, const float* wr_yfm,
    const float* b0_yom, const float* w1_yom,
    const float* b0_ylm, const float* w2_ylm,
    const float* w1_yum, const float* b0_yum)
{
  const int t = blockIdx.x * 256 + threadIdx.x;
  if (t >= NB) return;
  const Consts K = make_consts(cmean, cstd, wr_yom, wr_ylm, wr_yfm,
                               b0_yom, w1_yom, b0_ylm, w2_ylm, w1_yum, b0_yum);
  const float obsstd = stdp[0];
  const float mask = (t >= time_lag[0]) ? 1.0f : 0.0f;

  const float c  = cbuf[t];
  const float u1 = x[2 * t];
  const float u2 = x[2 * t + 1];

  float ib = 1.0f / (1.0f + exp2f(fmaf(c, K.kib_c, fmaf(u1, K.kib_u, K.kib_0))));
  float oo = K.oo1 / (1.0f + exp2f(fmaf(c, K.koo_c, K.koo_0)));
  float ol = K.ol1 / (1.0f + exp2f(fmaf(u2, K.kol_u, K.kol_0)));
  bool  pos   = c > 0.0f;
  float csafe = pos ? c : 1.0f;
  float olc   = pos ? (ol - fmaxf(0.0f, ol - u2 / csafe)) : ol;
  float f  = 1.0f - oo - olc;
  float bp = ib * u1;
  float h  = fmaf(oo, c, bp);

  out[            t] = h   * mask;   // h_n
  out[     NB  +  t] = c   * mask;   // c_n  (carry entering step t)
  out[2LL* NB  +  t] = ol  * c * mask;   // l_n
  out[3LL* NB  +  t] = olc * c * mask;   // lc_n
  out[4LL* NB  +  t] = bp  * mask;   // bp_n
  out[5LL* NB  +  t] = ib  * mask;   // g_ib
  out[6LL* NB  +  t] = oo  * mask;   // g_oo
  out[7LL* NB  +  t] = ol  * mask;   // g_ol
  out[8LL* NB  +  t] = olc * mask;   // g_olc
  out[9LL* NB  +  t] = f   * mask;   // g_f
  // h_nout = concat([h_n, obs_std], axis=1): row-interleaved (B,2)
  float2 hn = make_float2(h * mask, obsstd * mask);
  reinterpret_cast<float2*>(out + 10LL * NB)[t] = hn;
  out[12LL * NB + t] = obsstd * mask;  // obs_std
}

// -------------------------------------------------------------------------
// Launch: A (partial std) -> B (final std) -> S (serial scan) -> E (outputs)
// Workspace layout (bytes):
//   [0      ..  4095]  psum[1024]
//   [4096   ..  8191]  psq[1024]
//   [8192   ..  8195]  obsstd scalar
//   [16384  ..      ]  cbuf[NB]  (4 MB)
// -------------------------------------------------------------------------
extern "C" void kernel_launch(void* const* d_in, const int* in_sizes, int n_in,
                              void* d_out, int out_size, void* d_ws, size_t ws_size,
                              hipStream_t stream) {
  (void)in_sizes; (void)n_in; (void)out_size; (void)ws_size;
  const float* x        = (const float*)d_in[0];
  /* d_in[1] = epoch (unused) */
  const int*   time_lag = (const int*)  d_in[2];
  const float* y_obs    = (const float*)d_in[3];
  const float* cmean    = (const float*)d_in[4];
  const float* cstd     = (const float*)d_in[5];
  const float* wr_yom   = (const float*)d_in[6];
  const float* wr_ylm   = (const float*)d_in[7];
  const float* wr_yfm   = (const float*)d_in[8];
  const float* b0_yom   = (const float*)d_in[9];
  const float* w1_yom   = (const float*)d_in[10];
  const float* b0_ylm   = (const float*)d_in[11];
  const float* w2_ylm   = (const float*)d_in[12];
  const float* w1_yum   = (const float*)d_in[13];
  const float* b0_yum   = (const float*)d_in[14];

  float* psum = (float*)d_ws;
  float* psq  = psum + 1024;
  float* stdp = psum + 2048;
  float* cbuf = (float*)((char*)d_ws + 16384);
  float* out  = (float*)d_out;

  reduce_partial<<<dim3(1024), dim3(256), 0, stream>>>(y_obs, psum, psq);
  reduce_final  <<<dim3(1),    dim3(1024), 0, stream>>>(psum, psq, stdp);
  scan_kernel   <<<dim3(1),    dim3(32),  0, stream>>>(
      x, cbuf, cmean, cstd, wr_yom, wr_ylm, wr_yfm,
      b0_yom, w1_yom, b0_ylm, w2_ylm, w1_yum, b0_yum);
  out_kernel    <<<dim3((NB + 255) / 256), dim3(256), 0, stream>>>(
      x, cbuf, stdp, time_lag, out, cmean, cstd, wr_yom, wr_ylm, wr_yfm,
      b0_yom, w1_yom, b0_ylm, w2_ylm, w1_yum, b0_yum);
}